// FFMLayer_3367254360214
// MI455X (gfx1250) — compile-verified
//
#include <hip/hip_runtime.h>

typedef __attribute__((ext_vector_type(4))) float v4f;
typedef __attribute__((ext_vector_type(2))) float v2f;
typedef __attribute__((ext_vector_type(8))) float v8f;
typedef int v4i __attribute__((vector_size(16)));   // matches builtin's int4 param type

#define FF      39
#define KK      16
#define CELLS   (FF * FF)        // 1521 field pairs (incl. diagonal)
#define QUADS   (CELLS * 4)      // 6084 float4 per batch slice (97,344 B)
#define THREADS 256              // 8 wave32 per block

#define AS1 __attribute__((address_space(1)))
#define AS3 __attribute__((address_space(3)))

typedef AS1 v4i* gq_ptr;   // global int4*  (AS1 == HIP __device__)
typedef AS3 v4i* lq_ptr;   // LDS int4*     (AS3 == HIP __shared__)

#if __has_builtin(__builtin_amdgcn_global_load_async_to_lds_b128)
#define FFM_HAVE_ASYNC_LDS 1
#else
#define FFM_HAVE_ASYNC_LDS 0
#endif

__global__ __launch_bounds__(THREADS)
void ffm_pairwise_kernel(const float* __restrict__ in, float* __restrict__ out) {
    __shared__ v4f   tile[QUADS];          // one full (39,39,16) fp32 slice
    __shared__ float wsum[THREADS / 32];

    const int b   = blockIdx.x;
    const int tid = threadIdx.x;
    const v4f* src = (const v4f*)in + (size_t)b * QUADS;

    // ---- Stage the 97 KB slice into LDS.
    // Preferred: CDNA5 async memory->LDS DMA path (no VGPR round-trip, ASYNCcnt
    // tracked, waves keep issuing). Fallback: coalesced non-temporal b128 loads.
#if FFM_HAVE_ASYNC_LDS
    for (int u = tid; u < QUADS; u += THREADS) {
        gq_ptr g = (gq_ptr)(unsigned long long)(size_t)&src[u];
        lq_ptr l = (lq_ptr)(unsigned)(size_t)&tile[u];
        __builtin_amdgcn_global_load_async_to_lds_b128(g, l, /*offset=*/0, /*cpol=*/0);
    }
#if __has_builtin(__builtin_amdgcn_s_wait_asynccnt)
    __builtin_amdgcn_s_wait_asynccnt(0);
#else
    asm volatile("s_wait_asynccnt 0" ::: "memory");
#endif
#else
    for (int u = tid; u < QUADS; u += THREADS) {
        tile[u] = __builtin_nontemporal_load(&src[u]);
    }
#endif
    __syncthreads();

    // ---- Pairwise interaction out of LDS.
    // sum over ordered cells (i,j), i != j of dot(x[i,j,:], x[j,i,:]); scale 0.5 at end.
    // Unit u = cell*4 + quad -> primary stream is a linear LDS walk (conflict-free).
    float acc = 0.0f;
    for (int u = tid; u < QUADS; u += THREADS) {
        const int cell = u >> 2;
        const int q    = u & 3;
        const int i    = cell / FF;
        const int j    = cell - i * FF;
        const v4f x = tile[u];
        const v4f y = tile[(j * FF + i) * 4 + q];   // transpose partner
        const float d = x[0] * y[0] + x[1] * y[1] + x[2] * y[2] + x[3] * y[3];
        acc += (i != j) ? d : 0.0f;                  // branchless diagonal mask
    }

    // ---- Wave32 reduction on the matrix pipe:
    // A(16x4 f32) = lane partials in K=0 / K=2 slots, B = ones  =>  D[m][n] = p_m + p_{m+16}.
    // Column sums of D rows 0..7 (lanes 0-15) and rows 8..15 (lanes 16-31) cover all 32 lanes.
    v2f a;    a[0]    = acc;  a[1]    = 0.0f;
    v2f ones; ones[0] = 1.0f; ones[1] = 1.0f;
    v8f c = {};
    c = __builtin_amdgcn_wmma_f32_16x16x4_f32(
            /*neg_a=*/false, a, /*neg_b=*/false, ones,
            /*c_mod=*/(short)0, c, /*reuse_a=*/false, /*reuse_b=*/false);
    float s = c[0] + c[1] + c[2] + c[3] + c[4] + c[5] + c[6] + c[7];
    s += __shfl_xor(s, 16, 32);                     // combine lane halves -> full wave sum

    const int lane = tid & 31;
    const int wave = tid >> 5;
    if (lane == 0) wsum[wave] = s;
    __syncthreads();

    if (tid == 0) {
        float t = 0.0f;
        #pragma unroll
        for (int w = 0; w < THREADS / 32; ++w) t += wsum[w];
        out[b] = 0.5f * t;                          // ordered pairs counted twice
    }
}

extern "C" void kernel_launch(void* const* d_in, const int* in_sizes, int n_in,
                              void* d_out, int out_size, void* d_ws, size_t ws_size,
                              hipStream_t stream) {
    const float* in  = (const float*)d_in[0];
    float*       out = (float*)d_out;
    const int B = in_sizes[0] / (FF * FF * KK);     // 8192
    ffm_pairwise_kernel<<<dim3(B), dim3(THREADS), 0, stream>>>(in, out);
}